// MSELoss_68994354643177
// MI455X (gfx1250) — compile-verified
//
#include <hip/hip_runtime.h>

#define B_  16
#define H_  90
#define LP_ 512
#define K_  384
#define WL_ 65536   // W*L = 256*256

typedef __attribute__((ext_vector_type(2))) float v2f;
typedef __attribute__((ext_vector_type(8))) float v8f;

// ---------------------------------------------------------------------------
// Kernel 1: stable stream-compaction of mask rows.
// One block per batch item; 1024 threads; each thread owns a contiguous chunk
// of 64 mask entries so ascending order is preserved. LDS Hillis-Steele scan
// of per-thread one-counts gives each thread its output base.
// ---------------------------------------------------------------------------
__global__ void compact_mask_kernel(const int* __restrict__ mask,
                                    int* __restrict__ pos) {
    const int b = blockIdx.x;
    const int t = threadIdx.x;            // 0..1023
    const int CHUNK = WL_ / 1024;         // 64
    const int* mrow = mask + (size_t)b * WL_;

    int base_idx = t * CHUNK;
    int c = 0;
    #pragma unroll 4
    for (int i = 0; i < CHUNK; ++i) c += (mrow[base_idx + i] != 0);

    __shared__ int sc[1024];
    sc[t] = c;
    __syncthreads();
    // inclusive Hillis-Steele scan
    for (int off = 1; off < 1024; off <<= 1) {
        int v = (t >= off) ? sc[t - off] : 0;
        __syncthreads();
        sc[t] += v;
        __syncthreads();
    }
    int out = sc[t] - c;                  // exclusive prefix = write base

    int* prow = pos + b * K_;
    for (int i = 0; i < CHUNK; ++i) {
        if (mrow[base_idx + i] != 0) {
            if (out < K_) prow[out] = base_idx + i;
            ++out;
        }
    }
}

// ---------------------------------------------------------------------------
// Kernel 2: per-(b,h) gather + squared-difference partial sums.
// Grid = B*H blocks, 384 threads (12 waves). Deterministic fixed-order
// reduction: shfl_xor within wave, then thread 0 sums the 12 wave results.
// ---------------------------------------------------------------------------
__global__ void mse_partial_kernel(const float* __restrict__ cs,
                                   const float* __restrict__ cs_p,
                                   const int*   __restrict__ pos,
                                   float* __restrict__ partial) {
    const int blk = blockIdx.x;           // 0..B*H-1
    const int b = blk / H_;
    const int j = threadIdx.x;            // 0..383

    const int   pj   = pos[b * K_ + j];
    const float a    = cs[(size_t)blk * LP_ + j];
    const float p    = cs_p[(size_t)blk * WL_ + pj];
    const float d    = a - p;
    float v = d * d;

    // wave32 reduction (fixed order)
    #pragma unroll
    for (int off = 16; off > 0; off >>= 1) v += __shfl_xor(v, off, 32);

    __shared__ float wsum[12];
    if ((threadIdx.x & 31) == 0) wsum[threadIdx.x >> 5] = v;
    __syncthreads();
    if (threadIdx.x == 0) {
        float s = 0.0f;
        #pragma unroll
        for (int i = 0; i < 12; ++i) s += wsum[i];
        partial[blk] = s;
    }
}

// ---------------------------------------------------------------------------
// Kernel 3: single-wave final reduction using V_WMMA_F32_16X16X4_F32.
// Per-lane strided accumulation (fixed order), then the 32 lane partials are
// folded with one WMMA: A vgpr0 = s (=> A[m][0]=s[m], A[m][2]=s[m+16]),
// A vgpr1 = 0, B = all-ones (layout-independent). D row m == s[m]+s[m+16]
// broadcast across N, so summing the 8 D vgprs gives the low-half total in
// lanes 0-15 and the high-half total in lanes 16-31; one shfl_xor(16) add
// completes the exact f32 sum. EXEC is all-ones (full wave, no divergence).
// ---------------------------------------------------------------------------
__global__ void final_reduce_kernel(const float* __restrict__ partial,
                                    int n, float* __restrict__ out,
                                    float scale) {
    const int lane = threadIdx.x;         // 0..31, one wave
    float s = 0.0f;
    for (int i = lane; i < n; i += 32) s += partial[i];

    v2f a; a[0] = s;    a[1] = 0.0f;
    v2f bo; bo[0] = 1.0f; bo[1] = 1.0f;
    v8f c = {};
    c = __builtin_amdgcn_wmma_f32_16x16x4_f32(
            /*neg_a=*/false, a, /*neg_b=*/false, bo,
            /*c_mod=*/(short)0, c, /*reuse_a=*/false, /*reuse_b=*/false);

    float t = ((c[0] + c[1]) + (c[2] + c[3])) + ((c[4] + c[5]) + (c[6] + c[7]));
    t += __shfl_xor(t, 16, 32);
    if (lane == 0) out[0] = t * scale;
}

extern "C" void kernel_launch(void* const* d_in, const int* in_sizes, int n_in,
                              void* d_out, int out_size, void* d_ws, size_t ws_size,
                              hipStream_t stream) {
    const float* cs   = (const float*)d_in[0];   // (B,H,LP) f32
    const float* cs_p = (const float*)d_in[1];   // (B,H,W,L) f32
    const int*   mask = (const int*)d_in[2];     // (B,W,L) i32
    float* out = (float*)d_out;

    int*   pos     = (int*)d_ws;                              // B*K ints
    float* partial = (float*)((char*)d_ws + (size_t)B_ * K_ * sizeof(int));

    compact_mask_kernel<<<B_, 1024, 0, stream>>>(mask, pos);
    mse_partial_kernel<<<B_ * H_, K_, 0, stream>>>(cs, cs_p, pos, partial);
    final_reduce_kernel<<<1, 32, 0, stream>>>(partial, B_ * H_, out,
                                              1.0f / (float)(B_ * H_ * K_));
}